// VQVAEQuantizer_52862457479902
// MI455X (gfx1250) — compile-verified
//
#include <hip/hip_runtime.h>
#include <hip/hip_bf16.h>

typedef __attribute__((ext_vector_type(2))) float v2f;
typedef __attribute__((ext_vector_type(4))) float v4f;
typedef __attribute__((ext_vector_type(8))) float v8f;
typedef __attribute__((ext_vector_type(4))) unsigned int u32x4;
typedef __attribute__((ext_vector_type(8))) int i32x8;
typedef __attribute__((ext_vector_type(4))) int i32x4;

#define NUM_EMB   4096
#define EMB_DIM   256
#define IN_DIM    1024
#define COMMIT    0.25f
#define KCH       128     // K-chunk staged per TDM transfer in proj kernel

// ---------------------------------------------------------------------------
// Tensor Data Mover: DMA a [tile1 x tile0] f32 tile (row-major, line stride
// `stride0` elements) from global into LDS at byte offset `ldsoff`.
// D# per CDNA5 ISA §8.3-8.5: group0 {count=1, lds_addr, global_addr, type=2},
// group1 {data_size=4B, tensor_dim0/1, tile_dim0/1, dim0_stride}, groups 2/3
// zero (2-D tensor). Tracked by TENSORcnt.
// ---------------------------------------------------------------------------
__device__ __forceinline__ void tdm_load_2d(const void* gptr, unsigned ldsoff,
                                            unsigned dim0, unsigned dim1,
                                            unsigned stride0,
                                            unsigned tile0, unsigned tile1) {
  unsigned long long ga = (unsigned long long)gptr;
  u32x4 g0 = { 1u,                                  // count=1, no gather
               ldsoff,                              // lds_addr (bytes)
               (unsigned)ga,                        // global_addr[31:0]
               (unsigned)(ga >> 32) | (2u << 30) }; // global_addr[56:32] | type=2
  i32x8 g1 = { (int)0x00020000u,                    // data_size=2 (4B), mask=0
               (int)(dim0 << 16),                   // tensor_dim0[15:0]
               (int)((dim0 >> 16) | (dim1 << 16)),  // dim0[31:16] | dim1[15:0]
               (int)((dim1 >> 16) | (tile0 << 16)), // dim1[31:16] | tile_dim0
               (int)tile1,                          // tile_dim1 | tile_dim2=0
               (int)stride0,                        // dim0_stride[31:0]
               0, 0 };
  i32x4 z4 = { 0, 0, 0, 0 };
#if defined(__clang_major__) && __clang_major__ >= 23
  i32x8 z8 = { 0, 0, 0, 0, 0, 0, 0, 0 };
  __builtin_amdgcn_tensor_load_to_lds(g0, g1, z4, z4, z8, 0);
#else
  __builtin_amdgcn_tensor_load_to_lds(g0, g1, z4, z4, 0);
#endif
}

// Generic LDS pointer -> LDS byte offset (low 32 bits of the LDS aperture).
__device__ __forceinline__ unsigned lds_off(const void* p) {
  return (unsigned)(unsigned long long)p;
}

// ---------------------------------------------------------------------------
// Kernel 1: proj[N,256] = features[N,1024] @ W^T + b   (fp32 WMMA 16x16x4)
// grid = (N/16, 256/64), block = 128 (4 waves, one 16-col tile each).
// Features A-tile TDM-staged in 16xKCH double-buffered LDS chunks shared by
// all 4 waves; W fragments read direct (1 MB, L2-resident, reused 2048x).
// ---------------------------------------------------------------------------
__global__ void proj_kernel(const float* __restrict__ F,
                            const float* __restrict__ W,
                            const float* __restrict__ bias,
                            float* __restrict__ proj, int N) {
  __shared__ float aLds[2][16 * KCH];               // 2 x 8 KB

  const int lane  = threadIdx.x & 31;
  const int wave  = threadIdx.x >> 5;
  const int rowb  = blockIdx.x * 16;
  const int colb  = blockIdx.y * 64 + wave * 16;
  const int r     = lane & 15;
  const int khalf = (lane >> 4) << 1;               // 0 or 2

  if (wave == 0)
    tdm_load_2d(F + (size_t)rowb * IN_DIM, lds_off(&aLds[0][0]),
                IN_DIM, N, IN_DIM, KCH, 16);
  __builtin_amdgcn_s_wait_tensorcnt(0);
  __syncthreads();

  const float* brow = W + (size_t)(colb + r) * IN_DIM + khalf;
  v8f acc = {};

  for (int c = 0; c < IN_DIM / KCH; ++c) {
    if (wave == 0 && c + 1 < IN_DIM / KCH)
      tdm_load_2d(F + (size_t)rowb * IN_DIM + (c + 1) * KCH,
                  lds_off(&aLds[(c + 1) & 1][0]),
                  IN_DIM, N, IN_DIM, KCH, 16);

    const float* arow = &aLds[c & 1][r * KCH + khalf];
#pragma unroll
    for (int kb = 0; kb < KCH; kb += 4) {
      v2f a = *(const v2f*)(arow + kb);
      v2f b = *(const v2f*)(brow + c * KCH + kb);
      acc = __builtin_amdgcn_wmma_f32_16x16x4_f32(false, a, false, b,
                                                  (short)0, acc, false, false);
    }
    __builtin_amdgcn_s_wait_tensorcnt(0);
    __syncthreads();
  }

  const int crowb = rowb + ((lane >> 4) << 3);
  const int ccol  = colb + r;
  const float bv  = bias[ccol];
#pragma unroll
  for (int i = 0; i < 8; ++i)
    proj[(size_t)(crowb + i) * EMB_DIM + ccol] = acc[i] + bv;
}

// ---------------------------------------------------------------------------
// Kernel 2: cnorm[k] = sum(codebook[k,:]^2).  One wave per row.
// ---------------------------------------------------------------------------
__global__ void cnorm_kernel(const float* __restrict__ C,
                             float* __restrict__ cnorm) {
  const int lane = threadIdx.x & 31;
  const int wave = threadIdx.x >> 5;
  const int row  = blockIdx.x * 4 + wave;
  const float* p = C + (size_t)row * EMB_DIM + lane * 8;
  float s = 0.f;
#pragma unroll
  for (int i = 0; i < 8; ++i) { float v = p[i]; s += v * v; }
#pragma unroll
  for (int m = 16; m; m >>= 1) s += __shfl_xor(s, m, 32);
  if (lane == 0) cnorm[row] = s;
}

// ---------------------------------------------------------------------------
// Kernel 3: fused distance GEMM + argmin.
// score(n,k) = cnorm[k] - 2 * proj[n,:] . codebook[k,:]
// block = 64 (2 waves), grid = N/32. Each wave TDM-stages its own 16x256
// proj tile; the two waves share a double-buffered codebook tile that wave 0
// TDM-prefetches one tile ahead. All WMMA fragments come from LDS.
// ---------------------------------------------------------------------------
__global__ void argmin_kernel(const float* __restrict__ proj,
                              const float* __restrict__ CB,
                              const float* __restrict__ cnorm,
                              int* __restrict__ idx, int N) {
  __shared__ float pLds[2][16 * EMB_DIM];           // 32 KB (per-wave tiles)
  __shared__ float cLds[2][16 * EMB_DIM];           // 32 KB (shared, dbl-buf)

  const int lane  = threadIdx.x & 31;
  const int wave  = threadIdx.x >> 5;
  const int rowb  = (blockIdx.x * 2 + wave) * 16;
  const int r     = lane & 15;
  const int khalf = (lane >> 4) << 1;

  // Each wave DMA-stages its own proj tile; wave 0 also stages codebook tile 0.
  tdm_load_2d(proj + (size_t)rowb * EMB_DIM, lds_off(&pLds[wave][0]),
              EMB_DIM, N, EMB_DIM, EMB_DIM, 16);
  if (wave == 0)
    tdm_load_2d(CB, lds_off(&cLds[0][0]),
                EMB_DIM, NUM_EMB, EMB_DIM, EMB_DIM, 16);
  __builtin_amdgcn_s_wait_tensorcnt(0);
  __syncthreads();

  const float* arow = &pLds[wave][r * EMB_DIM + khalf];

  float bestv[8];
  int   besti[8];
#pragma unroll
  for (int i = 0; i < 8; ++i) { bestv[i] = 3.4e38f; besti[i] = 0; }

  for (int t = 0; t < NUM_EMB / 16; ++t) {
    if (wave == 0 && t + 1 < NUM_EMB / 16)
      tdm_load_2d(CB + (size_t)(t + 1) * 16 * EMB_DIM,
                  lds_off(&cLds[(t + 1) & 1][0]),
                  EMB_DIM, NUM_EMB, EMB_DIM, EMB_DIM, 16);

    const float* brow = &cLds[t & 1][r * EMB_DIM + khalf];
    v8f acc = {};
#pragma unroll 8
    for (int kb = 0; kb < EMB_DIM; kb += 4) {
      v2f a = *(const v2f*)(arow + kb);
      v2f b = *(const v2f*)(brow + kb);
      acc = __builtin_amdgcn_wmma_f32_16x16x4_f32(false, a, false, b,
                                                  (short)0, acc, false, false);
    }

    const int   col = t * 16 + r;
    const float cn  = cnorm[col];
#pragma unroll
    for (int i = 0; i < 8; ++i) {
      float score = cn - 2.0f * acc[i];
      if (score < bestv[i]) { bestv[i] = score; besti[i] = col; }
    }

    __builtin_amdgcn_s_wait_tensorcnt(0);
    __syncthreads();
  }

  // Reduce across the 16 lanes of each half (all hold the same row per VGPR i),
  // first-index tie-break to match jnp.argmin.
#pragma unroll
  for (int i = 0; i < 8; ++i) {
    float v  = bestv[i];
    int   ix = besti[i];
#pragma unroll
    for (int m = 1; m < 16; m <<= 1) {
      float ov = __shfl_xor(v, m, 32);
      int   oi = __shfl_xor(ix, m, 32);
      if (ov < v || (ov == v && oi < ix)) { v = ov; ix = oi; }
    }
    if ((lane & 15) == 0)
      idx[rowb + ((lane >> 4) << 3) + i] = ix;
  }
}

// ---------------------------------------------------------------------------
// Kernel 4: gather codebook[idx], write STE output, accumulate sq-err.
// ---------------------------------------------------------------------------
__global__ void zero_kernel(float* __restrict__ accum) { accum[0] = 0.f; }

__global__ void gather_loss_kernel(const float* __restrict__ proj,
                                   const float* __restrict__ CB,
                                   const int* __restrict__ idx,
                                   float* __restrict__ out_q,
                                   float* __restrict__ out_idxf,
                                   float* __restrict__ accum) {
  const int lane = threadIdx.x & 31;
  const int wave = threadIdx.x >> 5;
  const int row  = blockIdx.x * 8 + wave;
  const int ix   = idx[row];

  const float* q = CB    + (size_t)ix  * EMB_DIM + lane * 8;
  const float* p = proj  + (size_t)row * EMB_DIM + lane * 8;
  float*       o = out_q + (size_t)row * EMB_DIM + lane * 8;

  float s = 0.f;
#pragma unroll
  for (int i = 0; i < 8; ++i) {
    float pv = p[i], qv = q[i];
    float d  = qv - pv;
    o[i] = pv + d;                 // straight-through numerics
    s += d * d;
  }
#pragma unroll
  for (int m = 16; m; m >>= 1) s += __shfl_xor(s, m, 32);
  if (lane == 0) {
    atomicAdd(accum, s);
    out_idxf[row] = (float)ix;
  }
}

__global__ void finalize_kernel(const float* __restrict__ accum,
                                float* __restrict__ loss_out, int n) {
  loss_out[0] = (1.0f + COMMIT) * accum[0] / (float)((size_t)n * EMB_DIM);
}

// ---------------------------------------------------------------------------
extern "C" void kernel_launch(void* const* d_in, const int* in_sizes, int n_in,
                              void* d_out, int out_size, void* d_ws, size_t ws_size,
                              hipStream_t stream) {
  const float* F    = (const float*)d_in[0];   // [B,T,1024]
  const float* W    = (const float*)d_in[1];   // [256,1024]
  const float* bias = (const float*)d_in[2];   // [256]
  const float* CB   = (const float*)d_in[3];   // [4096,256]

  const int N = in_sizes[0] / IN_DIM;          // 32768 rows

  float* proj  = (float*)d_ws;                 // N*256 floats (32 MB)
  float* cnorm = proj + (size_t)N * EMB_DIM;   // 4096 floats
  int*   idx   = (int*)(cnorm + NUM_EMB);      // N ints
  float* accum = (float*)(idx + N);            // 1 float

  float* out_q    = (float*)d_out;             // N*256
  float* out_loss = out_q + (size_t)N * EMB_DIM;
  float* out_idxf = out_loss + 1;              // N

  proj_kernel<<<dim3(N / 16, EMB_DIM / 64), 128, 0, stream>>>(F, W, bias, proj, N);
  cnorm_kernel<<<NUM_EMB / 4, 128, 0, stream>>>(CB, cnorm);
  argmin_kernel<<<N / 32, 64, 0, stream>>>(proj, CB, cnorm, idx, N);
  zero_kernel<<<1, 1, 0, stream>>>(accum);
  gather_loss_kernel<<<N / 8, 256, 0, stream>>>(proj, CB, idx, out_q, out_idxf, accum);
  finalize_kernel<<<1, 1, 0, stream>>>(accum, out_loss, N);
}